// Tagger_40312563040572
// MI455X (gfx1250) — compile-verified
//
#include <hip/hip_runtime.h>
#include <math.h>

// Problem constants (from reference)
#define B_    16384
#define T_    99
#define E_    10
#define R_    10
#define L_    3
#define ROWS  (B_ * T_)           // 1622016

typedef __attribute__((ext_vector_type(2))) float v2f;
typedef __attribute__((ext_vector_type(8))) float v8f;

__device__ __forceinline__ float fast_tanh(float x) {
#if __has_builtin(__builtin_amdgcn_tanhf)
    return __builtin_amdgcn_tanhf(x);      // V_TANH_F32 (CDNA5 trans op)
#else
    return tanhf(x);
#endif
}

#define WAVES_P1  4
#define TPB_P1    (WAVES_P1 * 32)
#define TILES     (B_ / 16)              // 1024 waves total
#define BLOCKS_P1 (TILES / WAVES_P1)     // 256 blocks
#define HSTRIDE   20                     // LDS h-tile stride (floats); 16*20 banks conflict-free

// ---------------- Phase 0: pack tokens (2 bits each) per 16-row tile ----------------
// packed[tile][t] = sum_m token[tile*16+m][t] << (2*m)
__global__ __launch_bounds__(256) void pack_tokens_kernel(
    const int* __restrict__ tok, unsigned int* __restrict__ packed)
{
    int idx = blockIdx.x * 256 + threadIdx.x;     // TILES*T_ = 101376 exact
    int tile = idx / T_;
    int t    = idx % T_;
    unsigned int p = 0;
    #pragma unroll
    for (int m = 0; m < 16; ++m)
        p |= ((unsigned int)tok[(size_t)(tile * 16 + m) * T_ + t]) << (2 * m);
    packed[(size_t)tile * T_ + t] = p;
}

// ---------------- Phase 1: RNN recurrence via f32 WMMA ----------------
// One wave owns a 16-batch tile. h@U = three V_WMMA_F32_16X16X4_F32
// (K chunks 0-3,4-7,8-11). Pad lanes (n>=10) provably compute hv==0
// (zero U columns + zero xa => tanh(0)=0, carry of 0 stays 0), so the
// LDS writeback is unconditional and doubles as the K zero-padding.
__global__ __launch_bounds__(TPB_P1) void rnn_kernel(
    const unsigned int* __restrict__ packed, const float* __restrict__ emb,
    const float* __restrict__ W, const float* __restrict__ U,
    const float* __restrict__ bvec, float* __restrict__ states)
{
    __shared__ float hbuf[WAVES_P1][16 * HSTRIDE];   // per-wave h tile, [m][k]
    const int tid  = threadIdx.x;
    const int w    = tid >> 5;
    const int lane = tid & 31;
    const int n    = lane & 15;      // N column (rnn output j) in D layout
    const int half = lane >> 4;
    const int tileIdx = blockIdx.x * WAVES_P1 + w;
    const int batch0  = tileIdx * 16;
    float* hl = &hbuf[w][0];

    // zero the h tile (h0 = 0)
    #pragma unroll
    for (int i = lane; i < 16 * HSTRIDE; i += 32) hl[i] = 0.0f;

    // Per-lane pre-activation contribution for tokens 1 and 2:
    // a_t[n] = (emb[t] @ W)[n] + b[n].  Token 0 is masked (carry h).
    // Zero for pad lanes n>=10 (keeps pad hv exactly 0).
    float xa1 = 0.0f, xa2 = 0.0f;
    if (n < R_) {
        xa1 = bvec[n];
        xa2 = bvec[n];
        #pragma unroll
        for (int e = 0; e < E_; ++e) {
            float wv = W[e * R_ + n];
            xa1 += emb[1 * E_ + e] * wv;
            xa2 += emb[2 * E_ + e] * wv;
        }
    }

    // U in B-matrix layout (K x N), 3 chunks of K=4.
    // Mirrors documented f32 A layout: VGPR0 holds K = 4c + 2*half,
    // VGPR1 holds K = 4c + 2*half + 1. Out-of-range rows/cols -> 0.
    v2f ub[3];
    #pragma unroll
    for (int c = 0; c < 3; ++c) {
        int k0 = 4 * c + 2 * half;
        ub[c].x = (k0     < R_ && n < R_) ? U[(k0    ) * R_ + n] : 0.0f;
        ub[c].y = (k0 + 1 < R_ && n < R_) ? U[(k0 + 1) * R_ + n] : 0.0f;
    }

    const int shbase = 16 * half;    // token bit offset for this lane's D rows
    v8f hd = {};  // h carried in D layout (8 VGPRs): hd[r] = h[batch r+8*half][n]

    for (int t = 0; t < T_; ++t) {
        unsigned int pk = packed[(size_t)tileIdx * T_ + t];   // 16 tokens, 2b each

        // A-matrix chunks from LDS: a.x = h[m][4c+2*half], a.y = next k
        v2f av0, av1, av2;
        av0.x = hl[n * HSTRIDE + 0 + 2 * half];  av0.y = hl[n * HSTRIDE + 0 + 2 * half + 1];
        av1.x = hl[n * HSTRIDE + 4 + 2 * half];  av1.y = hl[n * HSTRIDE + 4 + 2 * half + 1];
        av2.x = hl[n * HSTRIDE + 8 + 2 * half];  av2.y = hl[n * HSTRIDE + 8 + 2 * half + 1];

        v8f d = {};
        d = __builtin_amdgcn_wmma_f32_16x16x4_f32(false, av0, false, ub[0], (short)0, d, false, false);
        d = __builtin_amdgcn_wmma_f32_16x16x4_f32(false, av1, false, ub[1], (short)0, d, false, false);
        d = __builtin_amdgcn_wmma_f32_16x16x4_f32(false, av2, false, ub[2], (short)0, d, false, false);

        #pragma unroll
        for (int r = 0; r < 8; ++r) {
            int m = r + 8 * half;                        // batch row for this D VGPR
            unsigned int tkm = (pk >> (shbase + 2 * r)) & 3u;
            float xw = (tkm == 1u) ? xa1 : xa2;
            float hn = fast_tanh(d[r] + xw);
            float hv = (tkm == 0u) ? hd[r] : hn;         // Keras masking: carry state
            hd[r] = hv;
            hl[m * HSTRIDE + n] = hv;                    // unconditional: pad lanes write 0
        }
        if (n < R_) {                                    // single exec toggle per timestep
            #pragma unroll
            for (int r = 0; r < 8; ++r)
                states[((size_t)(batch0 + r + 8 * half) * T_ + t) * R_ + n] = hd[r];
        }
    }
}

// ---------------- Phase 2: Dense + softmax + loss/accuracy partials ----------------
#define TPB_P2    256
#define BLOCKS_P2 (ROWS / TPB_P2)   // 6336 (exact)

__global__ __launch_bounds__(TPB_P2) void head_kernel(
    const float* __restrict__ states, const float* __restrict__ labels,
    const float* __restrict__ Wo, const float* __restrict__ bo,
    float* __restrict__ probs, float* __restrict__ partial)
{
    __shared__ float sW[R_ * L_ + L_];
    __shared__ float redl[TPB_P2];
    __shared__ float reda[TPB_P2];
    const int tid = threadIdx.x;
    if (tid < R_ * L_) sW[tid] = Wo[tid];
    if (tid < L_)      sW[R_ * L_ + tid] = bo[tid];
    __syncthreads();

    const size_t row = (size_t)blockIdx.x * TPB_P2 + tid;

    float h[R_];
    #pragma unroll
    for (int j = 0; j < R_; ++j) h[j] = states[row * R_ + j];

    float lg[L_];
    #pragma unroll
    for (int k = 0; k < L_; ++k) {
        float acc = sW[R_ * L_ + k];
        #pragma unroll
        for (int j = 0; j < R_; ++j) acc += h[j] * sW[j * L_ + k];
        lg[k] = acc;
    }
    float mx = fmaxf(lg[0], fmaxf(lg[1], lg[2]));
    float e0 = __expf(lg[0] - mx), e1 = __expf(lg[1] - mx), e2 = __expf(lg[2] - mx);
    float inv = 1.0f / (e0 + e1 + e2);
    float p[L_] = { e0 * inv, e1 * inv, e2 * inv };
    #pragma unroll
    for (int k = 0; k < L_; ++k) probs[row * L_ + k] = p[k];

    float lab[L_];
    #pragma unroll
    for (int k = 0; k < L_; ++k) lab[k] = labels[row * L_ + k];

    float loss_i = 0.0f;
    #pragma unroll
    for (int k = 0; k < L_; ++k) {
        float pc = fminf(fmaxf(p[k], 1e-7f), 1.0f);
        loss_i -= lab[k] * __logf(pc);
    }
    // first-occurrence argmax (matches jnp.argmax)
    int pi = 0; float pv = p[0];
    if (p[1] > pv) { pv = p[1]; pi = 1; }
    if (p[2] > pv) { pv = p[2]; pi = 2; }
    int li = 0; float lv = lab[0];
    if (lab[1] > lv) { lv = lab[1]; li = 1; }
    if (lab[2] > lv) { lv = lab[2]; li = 2; }
    float acc_i = (pi == li) ? 1.0f : 0.0f;

    redl[tid] = loss_i;
    reda[tid] = acc_i;
    __syncthreads();
    #pragma unroll
    for (int s = TPB_P2 / 2; s > 0; s >>= 1) {
        if (tid < s) { redl[tid] += redl[tid + s]; reda[tid] += reda[tid + s]; }
        __syncthreads();
    }
    if (tid == 0) {
        partial[blockIdx.x]             = redl[0];
        partial[BLOCKS_P2 + blockIdx.x] = reda[0];
    }
}

// ---------------- Phase 3: deterministic final reduction ----------------
__global__ __launch_bounds__(256) void finalize_kernel(
    const float* __restrict__ partial, float* __restrict__ out_scalars)
{
    __shared__ float redl[256];
    __shared__ float reda[256];
    const int tid = threadIdx.x;
    float sl = 0.0f, sa = 0.0f;
    for (int i = tid; i < BLOCKS_P2; i += 256) {   // fixed per-thread order
        sl += partial[i];
        sa += partial[BLOCKS_P2 + i];
    }
    redl[tid] = sl; reda[tid] = sa;
    __syncthreads();
    #pragma unroll
    for (int s = 128; s > 0; s >>= 1) {
        if (tid < s) { redl[tid] += redl[tid + s]; reda[tid] += reda[tid + s]; }
        __syncthreads();
    }
    if (tid == 0) {
        out_scalars[0] = reda[0] / (float)ROWS;   // accuracy
        out_scalars[1] = redl[0] / (float)ROWS;   // loss
    }
}

extern "C" void kernel_launch(void* const* d_in, const int* in_sizes, int n_in,
                              void* d_out, int out_size, void* d_ws, size_t ws_size,
                              hipStream_t stream)
{
    const int*   tok    = (const int*)d_in[0];
    const float* labels = (const float*)d_in[1];
    // d_in[2] = mask (unused by reference math)
    const float* emb    = (const float*)d_in[3];
    const float* W      = (const float*)d_in[4];
    const float* U      = (const float*)d_in[5];
    const float* bvec   = (const float*)d_in[6];
    const float* Wo     = (const float*)d_in[7];
    const float* bo     = (const float*)d_in[8];

    float* out     = (float*)d_out;
    float* states  = out;                                   // [B,T,10]
    float* probs   = out + (size_t)ROWS * R_;               // [B,T,3]
    float* scalars = out + (size_t)ROWS * (R_ + L_);        // accuracy, loss

    unsigned int* packed = (unsigned int*)d_ws;             // TILES*T_ u32 (~396 KB)
    float* partial = (float*)((char*)d_ws + (size_t)TILES * T_ * sizeof(unsigned int));

    pack_tokens_kernel<<<(TILES * T_) / 256, 256, 0, stream>>>(tok, packed);
    rnn_kernel<<<BLOCKS_P1, TPB_P1, 0, stream>>>(packed, emb, W, U, bvec, states);
    head_kernel<<<BLOCKS_P2, TPB_P2, 0, stream>>>(states, labels, Wo, bo, probs, partial);
    finalize_kernel<<<1, 256, 0, stream>>>(partial, scalars);
}